// MolTEGNNLayer_74113955660246
// MI455X (gfx1250) — compile-verified
//
#include <hip/hip_runtime.h>
#include <hip/hip_bf16.h>

#define NNODES 100000
#define NEDGES 600000
#define DIM    128
#define LDS_STRIDE 132   // 128 + 4 pad -> conflict-free row-major fragment reads
#define WAVES_PER_BLK 4
#define ROW_WAVES (NNODES / 16)   // 6250 waves of 16 rows each
#define WP_FRAGS (8 * 32 * 32)    // nt * kt * lane fragment-lanes per weight matrix

typedef __attribute__((ext_vector_type(2))) float v2f;
typedef __attribute__((ext_vector_type(8))) float v8f;

// ---------------------------------------------------------------------------
// Kernel 1: hpre = (1 + eps) * x   (pre-initializes the scatter accumulator)
// ---------------------------------------------------------------------------
__global__ __launch_bounds__(256) void gine_init_kernel(
    const float* __restrict__ x, const float* __restrict__ epsPtr,
    float* __restrict__ hpre)
{
    int i = blockIdx.x * blockDim.x + threadIdx.x;   // float4 index
    const int total = NNODES * DIM / 4;
    if (i >= total) return;
    float s = 1.0f + *epsPtr;
    float4 v = ((const float4*)x)[i];
    v.x *= s; v.y *= s; v.z *= s; v.w *= s;
    ((float4*)hpre)[i] = v;
}

// ---------------------------------------------------------------------------
// Kernel 1b: repack a 128x128 weight matrix into WMMA-B fragment order.
// Fragment (nt, kt, lane) = { W[k0][n], W[k0+1][n] } with
//   k0 = kt*4 + 2*(lane>=16), n = nt*16 + (lane&15)
// so the GEMM's B load becomes one coalesced global_load_b64 per lane.
// ---------------------------------------------------------------------------
__global__ __launch_bounds__(256) void pack_weights_kernel(
    const float* __restrict__ W, float* __restrict__ Wp)
{
    int t = blockIdx.x * blockDim.x + threadIdx.x;
    if (t >= WP_FRAGS) return;
    int lane = t & 31;
    int kt   = (t >> 5) & 31;
    int nt   = t >> 10;
    int k0 = kt * 4 + ((lane >> 4) << 1);
    int n  = nt * 16 + (lane & 15);
    Wp[2 * t]     = W[k0 * DIM + n];
    Wp[2 * t + 1] = W[(k0 + 1) * DIM + n];
}

// ---------------------------------------------------------------------------
// Kernel 2: per-edge message + scatter-add.
// One wave per edge; each lane handles 4 of the 128 features (float4).
// ---------------------------------------------------------------------------
__global__ __launch_bounds__(256) void gine_edge_kernel(
    const float* __restrict__ x, const int* __restrict__ ei,
    const float* __restrict__ ea, const float* __restrict__ We,
    const float* __restrict__ be, float* __restrict__ agg)
{
    int e = blockIdx.x * 8 + (threadIdx.x >> 5);
    if (e >= NEDGES) return;
    int lane = threadIdx.x & 31;

    int src = ei[e];             // edge_index row 0
    int dst = ei[NEDGES + e];    // edge_index row 1

    const float4* We4 = (const float4*)We;
    float4 emb = ((const float4*)be)[lane];
#pragma unroll
    for (int k = 0; k < 6; ++k) {
        float c = ea[e * 6 + k];          // wave-uniform -> scalar load
        float4 w = We4[k * 32 + lane];
        emb.x = fmaf(c, w.x, emb.x);
        emb.y = fmaf(c, w.y, emb.y);
        emb.z = fmaf(c, w.z, emb.z);
        emb.w = fmaf(c, w.w, emb.w);
    }
    float4 xv = ((const float4*)x)[src * 32 + lane];
    float4 m;
    m.x = fmaxf(xv.x + emb.x, 0.f);
    m.y = fmaxf(xv.y + emb.y, 0.f);
    m.z = fmaxf(xv.z + emb.z, 0.f);
    m.w = fmaxf(xv.w + emb.w, 0.f);

    float* p = agg + (size_t)dst * DIM + lane * 4;
    atomicAdd(p + 0, m.x);
    atomicAdd(p + 1, m.y);
    atomicAdd(p + 2, m.z);
    atomicAdd(p + 3, m.w);
}

// ---------------------------------------------------------------------------
// One MLP layer on a 16x128 tile in LDS using V_WMMA_F32_16X16X4_F32.
// kt-outer / nt-inner with 8 independent accumulators: each kt step issues
// 1 LDS b64 (A frag) + 8 coalesced global b64 (pre-packed B frags) + 8 WMMAs
// with RAW distance 8, letting the scheduler overlap loads with WMMA.
// ---------------------------------------------------------------------------
__device__ __forceinline__ void mlp_layer_wmma(
    const float* __restrict__ Ain, float* __restrict__ Out,
    const v2f* __restrict__ Wp, const float* __restrict__ Bias,
    int lane, bool do_relu)
{
    const int m    = lane & 15;
    const int koff = (lane >> 4) << 1;   // +2 for upper half-wave
    const int roff = (lane >> 4) << 3;   // +8 rows for upper half-wave

    v8f c[8];
#pragma unroll
    for (int nt = 0; nt < 8; ++nt) {
        float bias = Bias[nt * 16 + m];
#pragma unroll
        for (int g = 0; g < 8; ++g) c[nt][g] = bias;
    }

#pragma unroll 4
    for (int kt = 0; kt < 32; ++kt) {
        float2 af = *(const float2*)&Ain[m * LDS_STRIDE + kt * 4 + koff];
        v2f a; a.x = af.x; a.y = af.y;
#pragma unroll
        for (int nt = 0; nt < 8; ++nt) {
            v2f b = Wp[(unsigned)((nt * 32 + kt) * 32 + lane)];
            c[nt] = __builtin_amdgcn_wmma_f32_16x16x4_f32(
                        false, a, false, b, (short)0, c[nt], false, false);
        }
    }

#pragma unroll
    for (int nt = 0; nt < 8; ++nt) {
#pragma unroll
        for (int g = 0; g < 8; ++g) {
            float v = c[nt][g];
            if (do_relu) v = fmaxf(v, 0.f);
            Out[(g + roff) * LDS_STRIDE + nt * 16 + m] = v;
        }
    }
}

// ---------------------------------------------------------------------------
// Kernel 3: h = relu(hpre@W1+b1)@W2+b2 ; LayerNorm ; ReLU -> out
// One wave per 16-row tile, 4 waves per block, no cross-wave sync needed.
// ---------------------------------------------------------------------------
__global__ __launch_bounds__(128) void gine_mlp_ln_kernel(
    const float* __restrict__ hpre,
    const v2f* __restrict__ Wp1, const float* __restrict__ b1,
    const v2f* __restrict__ Wp2, const float* __restrict__ b2,
    const float* __restrict__ gamma, const float* __restrict__ beta,
    float* __restrict__ out)
{
    __shared__ float ldsA[WAVES_PER_BLK][16 * LDS_STRIDE];
    __shared__ float ldsI[WAVES_PER_BLK][16 * LDS_STRIDE];

    int w    = threadIdx.x >> 5;
    int lane = threadIdx.x & 31;
    int waveId = blockIdx.x * WAVES_PER_BLK + w;
    if (waveId >= ROW_WAVES) return;
    int row0 = waveId * 16;

    float* A = ldsA[w];
    float* I = ldsI[w];

    // Stage 16x128 A tile: coalesced 512B rows from global -> padded LDS
#pragma unroll 4
    for (int r = 0; r < 16; ++r) {
        float4 v = ((const float4*)hpre)[(size_t)(row0 + r) * 32 + lane];
        *(float4*)&A[r * LDS_STRIDE + lane * 4] = v;
    }

    // Layer 1 (ReLU) then Layer 2 (no activation), result back into A
    mlp_layer_wmma(A, I, Wp1, b1, lane, true);
    mlp_layer_wmma(I, A, Wp2, b2, lane, false);

    // LayerNorm + ReLU: 2 lanes per row, 64 cols each, combine via shfl_xor 16
    int r    = lane & 15;
    int half = lane >> 4;
    float s = 0.f, sq = 0.f;
#pragma unroll
    for (int j = 0; j < 64; j += 4) {
        float4 v = *(const float4*)&A[r * LDS_STRIDE + half * 64 + j];
        s  += v.x + v.y + v.z + v.w;
        sq += v.x * v.x + v.y * v.y + v.z * v.z + v.w * v.w;
    }
    s  += __shfl_xor(s, 16);
    sq += __shfl_xor(sq, 16);
    float mu   = s * (1.0f / DIM);
    float var  = sq * (1.0f / DIM) - mu * mu;
    float rstd = rsqrtf(var + 1e-5f);

#pragma unroll
    for (int j = 0; j < 64; j += 4) {
        int col = half * 64 + j;
        float4 v  = *(const float4*)&A[r * LDS_STRIDE + col];
        float4 g  = ((const float4*)gamma)[col >> 2];
        float4 bt = ((const float4*)beta)[col >> 2];
        float4 o;
        o.x = fmaxf((v.x - mu) * rstd * g.x + bt.x, 0.f);
        o.y = fmaxf((v.y - mu) * rstd * g.y + bt.y, 0.f);
        o.z = fmaxf((v.z - mu) * rstd * g.z + bt.z, 0.f);
        o.w = fmaxf((v.w - mu) * rstd * g.w + bt.w, 0.f);
        *(float4*)&out[(size_t)(row0 + r) * DIM + col] = o;
    }
}

// ---------------------------------------------------------------------------
extern "C" void kernel_launch(void* const* d_in, const int* in_sizes, int n_in,
                              void* d_out, int out_size, void* d_ws, size_t ws_size,
                              hipStream_t stream) {
    const float* x     = (const float*)d_in[0];
    const int*   ei    = (const int*)d_in[1];
    const float* ea    = (const float*)d_in[2];
    // d_in[3] = batch (unused: single graph)
    const float* We    = (const float*)d_in[4];
    const float* be    = (const float*)d_in[5];
    const float* eps   = (const float*)d_in[6];
    const float* W1    = (const float*)d_in[7];
    const float* b1    = (const float*)d_in[8];
    const float* W2    = (const float*)d_in[9];
    const float* b2    = (const float*)d_in[10];
    const float* gamma = (const float*)d_in[11];
    const float* beta  = (const float*)d_in[12];
    float* out  = (float*)d_out;

    // workspace layout: hpre (N*D floats) | Wp1 (2*WP_FRAGS) | Wp2 (2*WP_FRAGS)
    float* hpre = (float*)d_ws;
    float* Wp1  = hpre + (size_t)NNODES * DIM;
    float* Wp2  = Wp1 + 2 * WP_FRAGS;

    // 1) hpre = (1+eps)*x ; repack weights into WMMA fragment order
    {
        int total = NNODES * DIM / 4;
        gine_init_kernel<<<(total + 255) / 256, 256, 0, stream>>>(x, eps, hpre);
        pack_weights_kernel<<<(WP_FRAGS + 255) / 256, 256, 0, stream>>>(W1, Wp1);
        pack_weights_kernel<<<(WP_FRAGS + 255) / 256, 256, 0, stream>>>(W2, Wp2);
    }
    // 2) scatter-add edge messages into hpre
    {
        int blocks = (NEDGES + 7) / 8;   // 8 waves (edges) per 256-thread block
        gine_edge_kernel<<<blocks, 256, 0, stream>>>(x, ei, ea, We, be, hpre);
    }
    // 3) MLP (WMMA f32) + LayerNorm + ReLU
    {
        int blocks = (ROW_WAVES + WAVES_PER_BLK - 1) / WAVES_PER_BLK;
        gine_mlp_ln_kernel<<<blocks, 128, 0, stream>>>(
            hpre, (const v2f*)Wp1, b1, (const v2f*)Wp2, b2, gamma, beta, out);
    }
}